// CharRNN_68212670595982
// MI455X (gfx1250) — compile-verified
//
#include <hip/hip_runtime.h>
#include <math.h>

#define VOCAB 32000
#define BATCH 32
#define TSEQ  128
#define EDIM  256
#define HDIM  1024
#define G4H   4096            // 4*H
#define MROWS (BATCH * TSEQ)  // 4096

typedef __attribute__((ext_vector_type(2))) float v2f;
typedef __attribute__((ext_vector_type(8))) float v8f;

// ---- V_WMMA_F32_16X16X4_F32 operand loaders (wave32 lane layouts, ISA 7.12.2) ----
// A 16x4 (MxK): lanes 0-15 -> M=lane, lanes 16-31 -> M=lane-16; VGPR v holds K = (lane>=16)*2 + v
__device__ __forceinline__ v2f lda_tile(const float* __restrict__ A, int lda, int m0, int k0, int lane) {
  int m = m0 + (lane & 15);
  int k = k0 + ((lane >> 4) << 1);
  const float* p = A + (size_t)m * lda + k;
  v2f a; a.x = p[0]; a.y = p[1]; return a;
}
// B 4x16 (KxN) from a row-major [K x N] matrix: N = lane&15, K = (lane>=16)*2 + v
__device__ __forceinline__ v2f ldb_kn(const float* __restrict__ Bm, int ldb, int k0, int n0, int lane) {
  int n = n0 + (lane & 15);
  int k = k0 + ((lane >> 4) << 1);
  v2f b; b.x = Bm[(size_t)k * ldb + n]; b.y = Bm[(size_t)(k + 1) * ldb + n]; return b;
}
// B 4x16 (KxN) where memory is row-major [N x K] (i.e. B = W^T): each lane reads 2 floats of its row
__device__ __forceinline__ v2f ldb_nk(const float* __restrict__ Bm, int ldb, int k0, int n0, int lane) {
  int n = n0 + (lane & 15);
  int k = k0 + ((lane >> 4) << 1);
  const float* p = Bm + (size_t)n * ldb + k;
  v2f b; b.x = p[0]; b.y = p[1]; return b;
}
// C/D 16x16 f32: VGPR r -> M = r + (lane>=16)*8, N = lane&15
__device__ __forceinline__ void st_tile(float* __restrict__ C, int ldc, int m0, int n0, int lane, v8f c) {
  int n  = n0 + (lane & 15);
  int mb = m0 + ((lane >> 4) << 3);
#pragma unroll
  for (int r = 0; r < 8; ++r) C[(size_t)(mb + r) * ldc + n] = c[r];
}

#define WMMA4(c, a, b) \
  (c) = __builtin_amdgcn_wmma_f32_16x16x4_f32(false, (a), false, (b), (short)0, (c), false, false)

__device__ __forceinline__ float sigmoidf_(float x) { return 1.0f / (1.0f + expf(-x)); }

// ---------------- Kernel 1: embedding gather into X[t*B+b][E] ----------------
__global__ void k_embed(const int* __restrict__ ids, const float* __restrict__ emb,
                        float* __restrict__ X) {
  int r = blockIdx.x;            // r = t*BATCH + b
  int t = r >> 5, b = r & 31;
  int tok = ids[b * TSEQ + t];   // input_data is [B,T]
  const float4* src = (const float4*)(emb + (size_t)tok * EDIM);
  float4* dst = (float4*)(X + (size_t)r * EDIM);
  dst[threadIdx.x] = src[threadIdx.x];   // 64 threads * 16B = 256 floats
}

// ------- Kernel 2: hoisted input GEMM  Z1X[4096,4096] = X @ W1[0:E,:] + b1 -------
__global__ void k_zx(const float* __restrict__ X, const float* __restrict__ W1,
                     const float* __restrict__ b1, float* __restrict__ Z) {
  const int lane = threadIdx.x;
  const int m0 = blockIdx.x * 32;
  const int n0 = blockIdx.y * 16;
  float bias = b1[n0 + (lane & 15)];
  v8f c0, c1;
#pragma unroll
  for (int r = 0; r < 8; ++r) { c0[r] = bias; c1[r] = bias; }
  for (int k0 = 0; k0 < EDIM; k0 += 4) {
    v2f a0 = lda_tile(X, EDIM, m0, k0, lane);
    v2f a1 = lda_tile(X, EDIM, m0 + 16, k0, lane);
    v2f b  = ldb_kn(W1, G4H, k0, n0, lane);
    WMMA4(c0, a0, b);
    WMMA4(c1, a1, b);
  }
  st_tile(Z, G4H, m0, n0, lane, c0);
  st_tile(Z, G4H, m0 + 16, n0, lane, c1);
}

// ==== LSTM step kernels: 128 threads = 4 waves; K split across waves, partial z
// ==== accumulators reduced through LDS, gate epilogue parallel over all threads.
// LDS layout: zbuf[wave][gate][m(32)][n(16)]  => 4*4*32*16 floats = 32 KB
#define ZB(w, g, m, n) zbuf[((((w) * 4 + (g)) * 32 + (m)) * 16) + (n)]

// ------- Kernel 3: LSTM layer-1 step: z1 = Z1X[t] + h1_prev @ W1[E:,:]; gates -------
__global__ void k_lstm1(const float* __restrict__ Zx, const float* __restrict__ W1h,
                        const float* __restrict__ h1p, float* __restrict__ c1,
                        float* __restrict__ h1n) {
  __shared__ float zbuf[4 * 4 * 32 * 16];
  const int lane = threadIdx.x & 31;
  const int wave = threadIdx.x >> 5;       // 0..3, owns K chunk [wave*256, wave*256+256)
  const int n0 = blockIdx.x * 16;
  v8f acc[4][2];
#pragma unroll
  for (int g = 0; g < 4; ++g) {
    v8f z = {};
    acc[g][0] = z; acc[g][1] = z;
  }
  const int kb = wave * (HDIM / 4);
  for (int kk = 0; kk < HDIM / 4; kk += 4) {
    v2f a0 = lda_tile(h1p, HDIM, 0,  kb + kk, lane);   // A reused across all 4 gates
    v2f a1 = lda_tile(h1p, HDIM, 16, kb + kk, lane);
#pragma unroll
    for (int g = 0; g < 4; ++g) {
      v2f b = ldb_kn(W1h, G4H, kb + kk, g * HDIM + n0, lane);
      WMMA4(acc[g][0], a0, b);
      WMMA4(acc[g][1], a1, b);
    }
  }
  // scatter WMMA-layout accumulators into plain [g][m][n] LDS tiles
#pragma unroll
  for (int g = 0; g < 4; ++g)
#pragma unroll
    for (int half = 0; half < 2; ++half)
#pragma unroll
      for (int r = 0; r < 8; ++r) {
        int m = half * 16 + ((lane >> 4) << 3) + r;
        ZB(wave, g, m, lane & 15) = acc[g][half][r];
      }
  __syncthreads();
  // epilogue: 512 (m,n) elements over 128 threads
#pragma unroll
  for (int i = 0; i < 4; ++i) {
    int e = threadIdx.x + i * 128;
    int m = e >> 4, n = e & 15;
    float z[4];
#pragma unroll
    for (int g = 0; g < 4; ++g) {
      float s = Zx[(size_t)m * G4H + g * HDIM + n0 + n];  // x@W1x + b1 (hoisted)
      s += ZB(0, g, m, n) + ZB(1, g, m, n) + ZB(2, g, m, n) + ZB(3, g, m, n);
      z[g] = s;
    }
    float cp = c1[(size_t)m * HDIM + n0 + n];
    float cn = cp * sigmoidf_(z[2]) + sigmoidf_(z[0]) * tanhf(z[1]);
    float hn = tanhf(cn) * sigmoidf_(z[3]);
    c1[(size_t)m * HDIM + n0 + n]  = cn;   // owner-only: in-place safe
    h1n[(size_t)m * HDIM + n0 + n] = hn;   // double-buffered across t
  }
}

// ------- Kernel 4: LSTM layer-2 step: z2 = [h1,h2] @ W2 + b2; gates; emit HS row -------
__global__ void k_lstm2(const float* __restrict__ W2, const float* __restrict__ b2,
                        const float* __restrict__ h1c, const float* __restrict__ h2p,
                        float* __restrict__ c2, float* __restrict__ h2n,
                        float* __restrict__ HS, int t) {
  __shared__ float zbuf[4 * 4 * 32 * 16];
  const int lane = threadIdx.x & 31;
  const int wave = threadIdx.x >> 5;       // 0..3, owns K chunk [wave*512, wave*512+512)
  const int n0 = blockIdx.x * 16;
  v8f acc[4][2];
#pragma unroll
  for (int g = 0; g < 4; ++g) {
    v8f z = {};
    acc[g][0] = z; acc[g][1] = z;
  }
  const int kb = wave * 512;                       // row base in W2 (K = 2048)
  const float* Asrc = (wave < 2) ? h1c : h2p;      // first/second half of concat
  const int ab = (wave < 2) ? kb : kb - HDIM;      // row base within Asrc
  for (int kk = 0; kk < 512; kk += 4) {
    v2f a0 = lda_tile(Asrc, HDIM, 0,  ab + kk, lane);
    v2f a1 = lda_tile(Asrc, HDIM, 16, ab + kk, lane);
#pragma unroll
    for (int g = 0; g < 4; ++g) {
      v2f b = ldb_kn(W2, G4H, kb + kk, g * HDIM + n0, lane);
      WMMA4(acc[g][0], a0, b);
      WMMA4(acc[g][1], a1, b);
    }
  }
#pragma unroll
  for (int g = 0; g < 4; ++g)
#pragma unroll
    for (int half = 0; half < 2; ++half)
#pragma unroll
      for (int r = 0; r < 8; ++r) {
        int m = half * 16 + ((lane >> 4) << 3) + r;
        ZB(wave, g, m, lane & 15) = acc[g][half][r];
      }
  __syncthreads();
#pragma unroll
  for (int i = 0; i < 4; ++i) {
    int e = threadIdx.x + i * 128;
    int m = e >> 4, n = e & 15;      // m = batch index
    float z[4];
#pragma unroll
    for (int g = 0; g < 4; ++g) {
      float s = b2[g * HDIM + n0 + n];
      s += ZB(0, g, m, n) + ZB(1, g, m, n) + ZB(2, g, m, n) + ZB(3, g, m, n);
      z[g] = s;
    }
    float cp = c2[(size_t)m * HDIM + n0 + n];
    float cn = cp * sigmoidf_(z[2]) + sigmoidf_(z[0]) * tanhf(z[1]);
    float hn = tanhf(cn) * sigmoidf_(z[3]);
    c2[(size_t)m * HDIM + n0 + n]  = cn;
    h2n[(size_t)m * HDIM + n0 + n] = hn;
    HS[((size_t)m * TSEQ + t) * HDIM + n0 + n] = hn;   // output row order b*T + t
  }
}

// ------- Kernel 5: logits[4096,32000] = HS @ softmax_w^T + softmax_b -------
// Block = 4 waves stacked along M sharing the same 64 vocab columns (B reuse in WGP$).
// Per wave: 32x64 tile = 8 f32 accumulators; store-bandwidth-bound overall.
__global__ void k_proj(const float* __restrict__ HS, const float* __restrict__ SW,
                       const float* __restrict__ sb, float* __restrict__ out) {
  const int lane = threadIdx.x & 31;
  const int wave = threadIdx.x >> 5;                // 0..3
  const int m0 = blockIdx.x * 128 + wave * 32;
  const int v0 = blockIdx.y * 64;
  v8f acc[2][4];
#pragma unroll
  for (int j = 0; j < 4; ++j) {
    float bias = sb[v0 + j * 16 + (lane & 15)];
    v8f ci;
#pragma unroll
    for (int r = 0; r < 8; ++r) ci[r] = bias;
    acc[0][j] = ci; acc[1][j] = ci;
  }
  const float* wrow = SW + (size_t)(v0 + (lane & 15)) * HDIM;
  for (int k0 = 0; k0 < HDIM; k0 += 4) {
    __builtin_prefetch(wrow + k0 + 128, 0, 1);      // global_prefetch_b8 on the B rows
    v2f a0 = lda_tile(HS, HDIM, m0, k0, lane);
    v2f a1 = lda_tile(HS, HDIM, m0 + 16, k0, lane);
#pragma unroll
    for (int j = 0; j < 4; ++j) {
      v2f b = ldb_nk(SW, HDIM, k0, v0 + j * 16, lane);   // softmax_w is [V,H] -> B = W^T
      WMMA4(acc[0][j], a0, b);
      WMMA4(acc[1][j], a1, b);
    }
  }
#pragma unroll
  for (int j = 0; j < 4; ++j) {
    st_tile(out, VOCAB, m0,      v0 + j * 16, lane, acc[0][j]);
    st_tile(out, VOCAB, m0 + 16, v0 + j * 16, lane, acc[1][j]);
  }
}

extern "C" void kernel_launch(void* const* d_in, const int* in_sizes, int n_in,
                              void* d_out, int out_size, void* d_ws, size_t ws_size,
                              hipStream_t stream) {
  (void)in_sizes; (void)n_in; (void)out_size; (void)ws_size;
  const int*   ids = (const int*)d_in[0];
  const float* emb = (const float*)d_in[1];
  const float* W1  = (const float*)d_in[2];
  const float* b1  = (const float*)d_in[3];
  const float* W2  = (const float*)d_in[4];
  const float* b2  = (const float*)d_in[5];
  const float* sw  = (const float*)d_in[6];
  const float* sb  = (const float*)d_in[7];
  float* out = (float*)d_out;

  // workspace carve-out (~85 MB)
  char* p = (char*)d_ws;
  float* X   = (float*)p; p += (size_t)MROWS * EDIM * sizeof(float);   //  4 MB
  float* Z1X = (float*)p; p += (size_t)MROWS * G4H  * sizeof(float);   // 64 MB
  float* HS  = (float*)p; p += (size_t)MROWS * HDIM * sizeof(float);   // 16 MB
  const size_t SB = (size_t)BATCH * HDIM * sizeof(float);              // 128 KB each
  float* h1b[2]; float* h2b[2];
  h1b[0] = (float*)p; p += SB;  h1b[1] = (float*)p; p += SB;
  h2b[0] = (float*)p; p += SB;  h2b[1] = (float*)p; p += SB;
  float* c1 = (float*)p; p += SB;
  float* c2 = (float*)p; p += SB;

  // zero initial state (ws is poisoned, not zeroed)
  hipMemsetAsync(h1b[0], 0, SB, stream);
  hipMemsetAsync(h2b[0], 0, SB, stream);
  hipMemsetAsync(c1,     0, SB, stream);
  hipMemsetAsync(c2,     0, SB, stream);

  k_embed<<<MROWS, 64, 0, stream>>>(ids, emb, X);
  k_zx<<<dim3(MROWS / 32, G4H / 16), 32, 0, stream>>>(X, W1, b1, Z1X);

  const float* W1h = W1 + (size_t)EDIM * G4H;   // recurrent half of W1
  for (int t = 0; t < TSEQ; ++t) {
    const float* h1p = h1b[t & 1];  float* h1n = h1b[(t + 1) & 1];
    const float* h2p = h2b[t & 1];  float* h2n = h2b[(t + 1) & 1];
    k_lstm1<<<HDIM / 16, 128, 0, stream>>>(Z1X + (size_t)t * BATCH * G4H, W1h, h1p, c1, h1n);
    k_lstm2<<<HDIM / 16, 128, 0, stream>>>(W2, b2, h1n, h2p, c2, h2n, HS, t);
  }

  k_proj<<<dim3(MROWS / 128, VOCAB / 64), 128, 0, stream>>>(HS, sw, sb, out);
}